// DynamicFilter_29583734735117
// MI455X (gfx1250) — compile-verified
//
#include <hip/hip_runtime.h>
#include <cstdint>

#define IMG     256
#define C_CH    64
#define KK      5
#define HALO    2
#define TILE_H  16
#define TILE_W  64
#define ROWS    (TILE_H + 2*HALO)        // 20
#define COLS    (TILE_W + 2*HALO)        // 68
#define PITCH   72                       // padded row pitch in floats (16B-aligned windows)
#define TILE_FLOATS (ROWS * PITCH)       // 1440
#define TILE_BYTES  (TILE_FLOATS * 4)    // 5760
#define CHUNKS_PER_ROW (COLS / 2)        // 34 b64 chunks per row
#define CHUNKS_TOTAL   (ROWS * CHUNKS_PER_ROW) // 680
#define ASYNC_ITERS 3                    // ceil(680/256), wave-uniform

typedef float v2f __attribute__((ext_vector_type(2)));
typedef float v4f __attribute__((ext_vector_type(4)));

__global__ __launch_bounds__(256)
void dynfilter_kernel(const float* __restrict__ x,
                      const float* __restrict__ filters,
                      float* __restrict__ out)
{
    // two tile buffers + one 16B scratch slot for redirected (OOB/padding) async lanes
    __shared__ float smem[2 * TILE_FLOATS + 4];

    const int tid = threadIdx.x;
    const int b   = blockIdx.z;
    const int h0  = blockIdx.y * TILE_H;
    const int w0  = blockIdx.x * TILE_W;

    // gfx1250 flat encoding of an LDS pointer carries the LDS byte offset in low 32 bits
    const uint32_t lds_base    = (uint32_t)(uintptr_t)(&smem[0]);
    const uint32_t scratch_off = lds_base + (uint32_t)(2 * TILE_FLOATS * 4);

    // Zero both buffers once: halo slots outside the image stay zero for all channels.
    for (int i = tid; i < 2 * TILE_FLOATS + 4; i += 256) smem[i] = 0.0f;

    // Pixel ownership: row ty (0..15), 4 consecutive w at tx4; packed as two float2 pairs
    const int ty  = tid >> 4;
    const int tx4 = (tid & 15) << 2;
    const int h   = h0 + ty;

    // 25 filter taps for the 4 owned pixels, register-resident for all 64 channels
    v2f frA[KK * KK], frB[KK * KK];
    {
        const float* fb = filters + (((size_t)b * (KK*KK)) * IMG + (size_t)h) * IMG
                                  + (size_t)(w0 + tx4);
        #pragma unroll
        for (int p = 0; p < KK * KK; ++p) {
            v4f f = *reinterpret_cast<const v4f*>(fb + (size_t)p * IMG * IMG);
            frA[p] = (v2f){f.x, f.y};
            frB[p] = (v2f){f.z, f.w};
        }
    }

    const float* xplane0 = x + ((size_t)b * C_CH) * IMG * IMG;

    // ---- hoisted per-lane async copy setup (loop-invariant over channels) ----
    uint32_t     ldsoff[2][ASYNC_ITERS];
    const float* gsrc[ASYNC_ITERS];
    #pragma unroll
    for (int it = 0; it < ASYNC_ITERS; ++it) {
        int  t   = tid + it * 256;
        bool pad = (t >= CHUNKS_TOTAL);
        int  tt  = pad ? 0 : t;
        int  row = tt / CHUNKS_PER_ROW;
        int  j   = tt - row * CHUNKS_PER_ROW;
        int  gh  = h0 - HALO + row;
        int  gw  = w0 - HALO + 2 * j;
        bool inb = (!pad) & ((unsigned)gh < (unsigned)IMG)
                          & ((unsigned)gw < (unsigned)IMG);
        uint32_t rel = (uint32_t)(row * PITCH * 4 + j * 8);
        ldsoff[0][it] = inb ? (lds_base + rel) : scratch_off;
        ldsoff[1][it] = inb ? (lds_base + (uint32_t)TILE_BYTES + rel) : scratch_off;
        int cgh = gh < 0 ? 0 : (gh > IMG - 1 ? IMG - 1 : gh);
        int cgw = gw < 0 ? 0 : (gw > IMG - 2 ? IMG - 2 : gw);
        gsrc[it] = xplane0 + ((size_t)cgh * IMG + (size_t)cgw);
    }

    __syncthreads(); // zeros visible everywhere before any async LDS write

    // Exactly ASYNC_ITERS async instructions per wave per tile -> uniform ASYNCcnt
    auto issue = [&](int c, int buf) {
        #pragma unroll
        for (int it = 0; it < ASYNC_ITERS; ++it) {
            const float* g = gsrc[it] + (size_t)c * (IMG * IMG);
            asm volatile("global_load_async_to_lds_b64 %0, %1, off"
                         :: "v"(ldsoff[buf][it]), "v"(g)
                         : "memory");
        }
    };

    issue(0, 0); // prologue: tile for channel 0 in flight

    float* outb = out + (((size_t)b * C_CH) * IMG + (size_t)h) * IMG + (size_t)(w0 + tx4);

    // 5-tap sliding window over packed float2 pairs: windows 0..6, aligned ones free
    auto compute = [&](int c, const float* cur) {
        v2f accA = {0.f, 0.f};
        v2f accB = {0.f, 0.f};
        #pragma unroll
        for (int i = 0; i < KK; ++i) {
            const float* rowp = cur + (ty + i) * PITCH + tx4; // 16B aligned
            v4f lo = *reinterpret_cast<const v4f*>(rowp);
            v4f hi = *reinterpret_cast<const v4f*>(rowp + 4);
            v2f win[7];
            win[0] = (v2f){lo.x, lo.y};
            win[1] = (v2f){lo.y, lo.z};
            win[2] = (v2f){lo.z, lo.w};
            win[3] = (v2f){lo.w, hi.x};
            win[4] = (v2f){hi.x, hi.y};
            win[5] = (v2f){hi.y, hi.z};
            win[6] = (v2f){hi.z, hi.w};
            #pragma unroll
            for (int j = 0; j < KK; ++j) {
                // v_pk_fma_f32: pixels (w,w+1) and (w+2,w+3)
                accA = __builtin_elementwise_fma(win[j],     frA[i * KK + j], accA);
                accB = __builtin_elementwise_fma(win[j + 2], frB[i * KK + j], accB);
            }
        }
        v4f o = {accA.x, accA.y, accB.x, accB.y};
        // write-once output: bypass L2 residency to keep x halo lines hot
        __builtin_nontemporal_store(o, reinterpret_cast<v4f*>(outb + (size_t)c * IMG * IMG));
    };

    for (int c = 0; c < C_CH; c += 2) {
        // ---- even channel (buffer 0) ----
        issue(c + 1, 1);                                    // prefetch odd channel
        asm volatile("s_wait_asynccnt 3" ::: "memory");     // tile c complete
        __syncthreads();
        compute(c, smem);
        __syncthreads();                                    // all reads of buf0 done

        // ---- odd channel (buffer 1) ----
        if (c + 2 < C_CH) {
            issue(c + 2, 0);                                // prefetch next even channel
            asm volatile("s_wait_asynccnt 3" ::: "memory"); // tile c+1 complete
        } else {
            asm volatile("s_wait_asynccnt 0" ::: "memory"); // final tile complete
        }
        __syncthreads();
        compute(c + 1, smem + TILE_FLOATS);
        __syncthreads();                                    // all reads of buf1 done
    }
}

extern "C" void kernel_launch(void* const* d_in, const int* in_sizes, int n_in,
                              void* d_out, int out_size, void* d_ws, size_t ws_size,
                              hipStream_t stream) {
    const float* x       = (const float*)d_in[0];
    const float* filters = (const float*)d_in[1];
    float*       out     = (float*)d_out;
    // B=8, C=64, H=W=256, K=5 per reference setup_inputs()
    dim3 grid(IMG / TILE_W, IMG / TILE_H, 8);
    dim3 block(256);
    dynfilter_kernel<<<grid, block, 0, stream>>>(x, filters, out);
}